// Embedding_3221225472252
// MI455X (gfx1250) — compile-verified
//
#include <hip/hip_runtime.h>

typedef __attribute__((ext_vector_type(2))) float v2f;
typedef __attribute__((ext_vector_type(4))) float v4f;
typedef __attribute__((ext_vector_type(8))) float v8f;

#define N_TOK    131072            // 32*64*64 tokens
#define K_CODES  512
#define D_DIM    64
#define HW       4096              // 64*64 spatial positions per image
#define B_STRIDE (64*4096)         // elements per image in NCHW
#define N_ELEM   (32*64*64*64)     // 8388608 elements of x / quant output

#define MT          2              // row-tiles (of 16) per wave
#define WPB         8              // waves per block
#define ROWS_PER_BLOCK (WPB * MT * 16)   // 256
#define STAGE_CODES 128            // codes staged in LDS per stage
#define LDS_STRIDE  68             // padded row stride (68 mod 64 = 4 -> conflict-free)

// ---------------------------------------------------------------------------
// K0: per-code squared norms  ||e_k||^2  -> ws
// ---------------------------------------------------------------------------
__global__ void code_norm_kernel(const float* __restrict__ w,
                                 float* __restrict__ wsq) {
    const int c = blockIdx.x * 256 + threadIdx.x;
    if (c < K_CODES) {
        const float* wr = w + c * D_DIM;
        float s = 0.0f;
#pragma unroll
        for (int d = 0; d < D_DIM; d += 4) {
            v4f v = *(const v4f*)(wr + d);
            s += v.x * v.x + v.y * v.y + v.z * v.z + v.w * v.w;
        }
        wsq[c] = s;
    }
}

// ---------------------------------------------------------------------------
// K1: WMMA fp32 GEMM (x . e) + running argmin over ||e||^2 - 2 x.e
// Block = 8 waves; wave handles MT=2 row-tiles (32 rows) vs all 512 codes.
// Codebook staged through LDS in 4 chunks of 128 codes (padded stride 68
// floats -> bank-conflict-free ds_load_b64 B-fragment reads).
// A fragment (16x4 f32): lane%16 = row M, lane/16 selects K-pair, vgpr = K%2
// B fragment (4x16 f32): lane%16 = col N, lane/16 selects K-pair, vgpr = K%2
// C fragment: vgpr r -> M = r + 8*(lane/16), N = lane%16
// ---------------------------------------------------------------------------
__global__ void __launch_bounds__(256)
vq_argmin_kernel(const float* __restrict__ x,
                 const float* __restrict__ w,
                 const float* __restrict__ wsq,
                 float* __restrict__ out_idx) {
    __shared__ float lw[STAGE_CODES * LDS_STRIDE];   // ~34 KB

    const int tid    = threadIdx.x;
    const int lane   = tid & 31;
    const int waveid = tid >> 5;
    const int half   = lane >> 4;
    const int l16    = lane & 15;
    const int gw     = blockIdx.x * WPB + waveid;

    // Resident A fragments for both row-tiles (reused across all 512 codes).
    // Tiles of 16 consecutive tokens never cross an image (4096 % 32 == 0).
    v2f a[MT][16];
#pragma unroll
    for (int t = 0; t < MT; ++t) {
        const int m0 = gw * (MT * 16) + t * 16;
        const float* xc = x + (m0 >> 12) * B_STRIDE + (m0 & 4095) + l16;
#pragma unroll
        for (int dk = 0; dk < 16; ++dk) {
            a[t][dk].x = xc[(4 * dk + 2 * half) * HW];
            a[t][dk].y = xc[(4 * dk + 2 * half + 1) * HW];
        }
    }

    float best[MT][8];
    int   bidx[MT][8];
#pragma unroll
    for (int t = 0; t < MT; ++t)
#pragma unroll
        for (int r = 0; r < 8; ++r) { best[t][r] = 3.0e38f; bidx[t][r] = 0; }

    for (int ks = 0; ks < K_CODES / STAGE_CODES; ++ks) {
        __syncthreads();
        // Cooperative stage: 128 codes x 64 floats, coalesced global reads.
        const float* wsrc = w + ks * STAGE_CODES * D_DIM;
#pragma unroll
        for (int i = 0; i < STAGE_CODES * D_DIM / 256; ++i) {
            const int lin = tid + i * 256;
            lw[(lin >> 6) * LDS_STRIDE + (lin & 63)] = wsrc[lin];
        }
        __syncthreads();

        for (int kt = 0; kt < STAGE_CODES / 16; ++kt) {
            const int code = ks * STAGE_CODES + kt * 16 + l16;
            const float* lb = lw + (kt * 16 + l16) * LDS_STRIDE + 2 * half;
            v8f c0 = {}, c1 = {};
#pragma unroll
            for (int dk = 0; dk < 16; ++dk) {
                const v2f b = *(const v2f*)(lb + 4 * dk);  // ds_load_b64
                c0 = __builtin_amdgcn_wmma_f32_16x16x4_f32(
                        false, a[0][dk], false, b, (short)0, c0, false, false);
                c1 = __builtin_amdgcn_wmma_f32_16x16x4_f32(
                        false, a[1][dk], false, b, (short)0, c1, false, false);
            }
            // dist surrogate: ||e||^2 - 2 x.e   (||x||^2 is row-constant)
            const float wq = wsq[code];
#pragma unroll
            for (int r = 0; r < 8; ++r) {
                const float d0 = wq - 2.0f * c0[r];
                if (d0 < best[0][r]) { best[0][r] = d0; bidx[0][r] = code; }
                const float d1 = wq - 2.0f * c1[r];
                if (d1 < best[1][r]) { best[1][r] = d1; bidx[1][r] = code; }
            }
        }
    }

    // Cross-lane argmin over the 16 code columns (stays within each half).
#pragma unroll
    for (int t = 0; t < MT; ++t) {
#pragma unroll
        for (int r = 0; r < 8; ++r) {
#pragma unroll
            for (int off = 1; off < 16; off <<= 1) {
                const float ov = __shfl_xor(best[t][r], off, 32);
                const int   oi = __shfl_xor(bidx[t][r], off, 32);
                if (ov < best[t][r] || (ov == best[t][r] && oi < bidx[t][r])) {
                    best[t][r] = ov; bidx[t][r] = oi;
                }
            }
        }
    }

    if (l16 == 0) {
#pragma unroll
        for (int t = 0; t < MT; ++t) {
            const int m0 = gw * (MT * 16) + t * 16;
#pragma unroll
            for (int r = 0; r < 8; ++r)
                out_idx[m0 + 8 * half + r] = (float)bidx[t][r];  // exact in fp32
        }
    }
}

// ---------------------------------------------------------------------------
// K2: quant_st (== quantized numerically) in NCHW + per-block loss partials
// ---------------------------------------------------------------------------
__global__ void quant_loss_kernel(const float* __restrict__ x,
                                  const float* __restrict__ w,
                                  const float* __restrict__ idxf,
                                  float* __restrict__ out_quant,
                                  float* __restrict__ partials) {
    __shared__ float sdata[256];
    const int e   = blockIdx.x * 256 + threadIdx.x;   // NCHW element
    const int b   = e >> 18;                          // / B_STRIDE
    const int rem = e & (B_STRIDE - 1);
    const int d   = rem >> 12;                        // channel
    const int p   = rem & 4095;                       // spatial position
    const int n   = (b << 12) + p;                    // token index
    const int k   = (int)idxf[n];
    const float q    = w[k * D_DIM + d];
    const float diff = q - x[e];
    __builtin_nontemporal_store(q, out_quant + e);

    sdata[threadIdx.x] = diff * diff;
    __syncthreads();
    for (int s = 128; s > 0; s >>= 1) {
        if (threadIdx.x < s) sdata[threadIdx.x] += sdata[threadIdx.x + s];
        __syncthreads();
    }
    if (threadIdx.x == 0) partials[blockIdx.x] = sdata[0];
}

// ---------------------------------------------------------------------------
// K3: one-hot enc [N_TOK, 512] — 268 MB of pure streaming NT stores (dominant)
// ---------------------------------------------------------------------------
__global__ void enc_kernel(const float* __restrict__ idxf,
                           float* __restrict__ out_enc) {
    const int g = blockIdx.x * 256 + threadIdx.x;     // float4 index
    const int n = g >> 7;                             // 128 float4 per row
    const int j = g & 127;
    const int k = (int)idxf[n];
    v4f v = {0.0f, 0.0f, 0.0f, 0.0f};
    if ((k >> 2) == j) v[k & 3] = 1.0f;
    __builtin_nontemporal_store(v, (v4f*)out_enc + g);
}

// ---------------------------------------------------------------------------
// K4: deterministic fixed-order loss finalize: loss = 2 * mean(diff^2)
// ---------------------------------------------------------------------------
__global__ void loss_final_kernel(const float* __restrict__ partials,
                                  float* __restrict__ out_loss, int nparts) {
    __shared__ float sdata[256];
    float s = 0.0f;
    for (int i = threadIdx.x; i < nparts; i += 256) s += partials[i];
    sdata[threadIdx.x] = s;
    __syncthreads();
    for (int st = 128; st > 0; st >>= 1) {
        if (threadIdx.x < st) sdata[threadIdx.x] += sdata[threadIdx.x + st];
        __syncthreads();
    }
    if (threadIdx.x == 0) out_loss[0] = 2.0f * sdata[0] / (float)N_ELEM;
}

// ---------------------------------------------------------------------------
extern "C" void kernel_launch(void* const* d_in, const int* in_sizes, int n_in,
                              void* d_out, int out_size, void* d_ws, size_t ws_size,
                              hipStream_t stream) {
    const float* x = (const float*)d_in[0];   // [32,64,64,64] NCHW fp32
    const float* w = (const float*)d_in[1];   // [512,64] fp32

    float* out   = (float*)d_out;
    float* loss  = out;                                       // [1]
    float* quant = out + 1;                                   // [8388608]
    float* enc   = quant + N_ELEM;                            // [67108864]
    float* idxf  = enc + (size_t)N_TOK * K_CODES;             // [131072]

    float* wsq      = (float*)d_ws;           // 512 floats
    float* partials = wsq + K_CODES;          // 32768 floats

    code_norm_kernel<<<2, 256, 0, stream>>>(w, wsq);

    // 512 blocks x 8 waves; each wave owns 32 rows vs all 512 codes
    vq_argmin_kernel<<<N_TOK / ROWS_PER_BLOCK, 256, 0, stream>>>(x, w, wsq, idxf);

    quant_loss_kernel<<<N_ELEM / 256, 256, 0, stream>>>(x, w, idxf, quant, partials);

    enc_kernel<<<((size_t)N_TOK * K_CODES / 4) / 256, 256, 0, stream>>>(idxf, enc);

    loss_final_kernel<<<1, 256, 0, stream>>>(partials, loss, N_ELEM / 256);
}